// StreamingDurationProjector_15814069584475
// MI455X (gfx1250) — compile-verified
//
#include <hip/hip_runtime.h>

// StreamingDurationProjector for MI455X (gfx1250, wave32).
// One wave32 workgroup per batch row; double-buffered async global->LDS tile DMA
// (ASYNCcnt), serial floor-carry recurrence confined to the window [P, L),
// b128-vectorized, non-temporal output production.

#define LANES  32
#define VEC    4
#define TILE   256                      // floats per tile
#define CHUNKS (TILE / (LANES * VEC))   // 2 b128 chunks per plane per tile
#define NLOADS (2 * CHUNKS)             // async loads in flight per tile (2 planes)

#if defined(__HIP_DEVICE_COMPILE__)
#if __has_builtin(__builtin_amdgcn_global_load_async_to_lds_b128)
#define HAVE_ASYNC_B128 1
#else
#define HAVE_ASYNC_B128 0
#endif
#if __has_builtin(__builtin_amdgcn_s_wait_asynccnt)
#define HAVE_WAIT_ASYNC 1
#else
#define HAVE_WAIT_ASYNC 0
#endif
#else
#define HAVE_ASYNC_B128 0
#define HAVE_WAIT_ASYNC 0
#endif

typedef float float4v __attribute__((ext_vector_type(4)));
typedef int   int4v   __attribute__((ext_vector_type(4)));
#if defined(__HIP_DEVICE_COMPILE__)
// Async builtin signature (probe-confirmed): (AS1 int4*, AS3 ptr, imm off, imm cpol).
typedef __attribute__((address_space(1))) int4v gv4i_t;
typedef __attribute__((address_space(3))) int4v lv4i_t;
#endif

__global__ __launch_bounds__(LANES)
void sdp_scan_kernel(const float* __restrict__ dur,
                     const float* __restrict__ spe,
                     const float* __restrict__ resid,
                     const float* __restrict__ cacheprev,
                     const int*   __restrict__ ulen,
                     const int*   __restrict__ slen,
                     const int*   __restrict__ cprev,
                     float* __restrict__ out_mat,
                     float* __restrict__ out_proj,
                     float* __restrict__ out_res,
                     float* __restrict__ out_cached,
                     int*   __restrict__ out_comm,
                     int U)
{
    __shared__ float s_dur[2][TILE];
    __shared__ float s_aux[2][TILE];    // speech mask in; frames out (in-place)

    const int b    = blockIdx.x;
    const int lane = threadIdx.x;

    const int L = min(ulen[b], slen[b]);            // committed prefix length
    int P = cprev[b];
    P = P < 0 ? 0 : (P > U ? U : P);                // cached prefix length

    const float* rdur   = dur       + (size_t)b * U;
    const float* rspe   = spe       + (size_t)b * U;
    const float* rcache = cacheprev + (size_t)b * U;
    const size_t rowbase = (size_t)b * U;

    float carry = resid[b];

    auto issue_tile = [&](int t0, int buf) {
#if HAVE_ASYNC_B128
        #pragma unroll
        for (int c = 0; c < CHUNKS; ++c) {
            const int off = c * LANES * VEC + lane * VEC;
            __builtin_amdgcn_global_load_async_to_lds_b128(
                (gv4i_t*)(rdur + t0 + off), (lv4i_t*)&s_dur[buf][off], 0, 0);
            __builtin_amdgcn_global_load_async_to_lds_b128(
                (gv4i_t*)(rspe + t0 + off), (lv4i_t*)&s_aux[buf][off], 0, 0);
        }
#else
        #pragma unroll
        for (int c = 0; c < CHUNKS; ++c) {
            const int g = c * LANES + lane;
            ((float4v*)s_dur[buf])[g] = ((const float4v*)(rdur + t0))[g];
            ((float4v*)s_aux[buf])[g] = ((const float4v*)(rspe + t0))[g];
        }
#endif
    };

    issue_tile(0, 0);

    const int ntiles = U / TILE;
    for (int t = 0; t < ntiles; ++t) {
        const int t0  = t * TILE;
        const int buf = t & 1;

        if (t + 1 < ntiles) {
            issue_tile(t0 + TILE, buf ^ 1);         // overlap DMA of next tile
#if HAVE_ASYNC_B128
#if HAVE_WAIT_ASYNC
            __builtin_amdgcn_s_wait_asynccnt(NLOADS);   // tile t done; t+1 in flight
#else
            asm volatile("s_wait_asynccnt %0" :: "i"(NLOADS) : "memory");
#endif
#endif
        } else {
#if HAVE_ASYNC_B128
#if HAVE_WAIT_ASYNC
            __builtin_amdgcn_s_wait_asynccnt(0);
#else
            asm volatile("s_wait_asynccnt 0" ::: "memory");
#endif
#endif
        }
        asm volatile("" ::: "memory");

        // Prefetch the cached-prefix stream this tile will read (global_prefetch_b8).
        if (t0 < P)
            __builtin_prefetch(rcache + t0 + lane * (TILE / LANES), 0, 0);

        __syncthreads();

        float* sd = s_dur[buf];
        float* sa = s_aux[buf];

        // Serial recurrence only where a real data dependency exists: [P, L) ∩ tile.
        if (lane == 0 && t0 < L && t0 + TILE > P) {
            int lo = P - t0; if (lo < 0) lo = 0;
            int hi = L - t0; if (hi > TILE) hi = TILE;
            float car = carry;
            auto step = [&](float d, float sp) -> float {
                float total = fmaxf(0.0f, d + car);
                float f = floorf(total);
                if (sp > 0.5f) f = fmaxf(1.0f, f);
                car = total - f;
                return f;
            };
            int i = lo;
            while (i < hi && (i & 3)) { sa[i] = step(sd[i], sa[i]); ++i; }
            for (; i + VEC <= hi; i += VEC) {
                float4v d4 = *(const float4v*)&sd[i];   // one ds_load_b128
                float4v a4 = *(const float4v*)&sa[i];
                float4v f4;
                #pragma unroll
                for (int k = 0; k < VEC; ++k) f4[k] = step(d4[k], a4[k]);
                *(float4v*)&sa[i] = f4;                 // one ds_store_b128
            }
            for (; i < hi; ++i) sa[i] = step(sd[i], sa[i]);
            carry = car;
        }
        __syncthreads();

        // Elementwise production of all three [B,U] outputs, b128 per lane, coalesced.
        // Output streams are write-once -> non-temporal so they don't evict the
        // read-once input streams from L2.
        #pragma unroll
        for (int c = 0; c < CHUNKS; ++c) {
            const int ibase = c * LANES * VEC + lane * VEC;
            const int ubase = t0 + ibase;
            const float4v d4 = ((const float4v*)sd)[c * LANES + lane];
            const float4v a4 = ((const float4v*)sa)[c * LANES + lane];

            float4v cache4 = {0.0f, 0.0f, 0.0f, 0.0f};
            const bool group_cached = (ubase + VEC <= P);
            if (group_cached)       // read-once stream: non-temporal b128 load
                cache4 = __builtin_nontemporal_load((const float4v*)(rcache + ubase));

            float4v m4, p4, c4;
            #pragma unroll
            for (int k = 0; k < VEC; ++k) {
                const int u = ubase + k;
                float proj;
                if (u < P)       proj = group_cached ? cache4[k] : rcache[u];
                else if (u < L)  proj = a4[k];       // freshly scanned frames
                else             proj = 0.0f;        // beyond commit: scanned == 0
                const float commitf = (u < L) ? 1.0f : 0.0f;
                const float pc = proj * commitf;
                m4[k] = d4[k] + (pc - d4[k]);        // JAX straight-through form
                p4[k] = proj;
                c4[k] = pc;
            }
            __builtin_nontemporal_store(m4, (float4v*)(out_mat    + rowbase + ubase));
            __builtin_nontemporal_store(p4, (float4v*)(out_proj   + rowbase + ubase));
            __builtin_nontemporal_store(c4, (float4v*)(out_cached + rowbase + ubase));
        }
        __syncthreads();                 // all lanes done with buf before its reuse
    }

    if (lane == 0) {
        out_res[b]  = carry;             // residual_next
        out_comm[b] = L;                 // commit_mask.sum() == min(ul, sl)
    }
}

extern "C" void kernel_launch(void* const* d_in, const int* in_sizes, int n_in,
                              void* d_out, int out_size, void* d_ws, size_t ws_size,
                              hipStream_t stream) {
    const float* dur    = (const float*)d_in[0];   // [B,U]
    const float* spe    = (const float*)d_in[1];   // [B,U]
    const float* resid  = (const float*)d_in[2];   // [B,1]
    const float* cacheA = (const float*)d_in[3];   // [B,U]
    const int*   ulen   = (const int*)d_in[4];     // [B]
    const int*   slen   = (const int*)d_in[5];     // [B]
    const int*   cprev  = (const int*)d_in[6];     // [B]

    const int B = in_sizes[4];
    const int U = in_sizes[0] / B;                 // 2048, 4096

    float* out = (float*)d_out;
    const size_t BU = (size_t)B * U;
    float* out_mat    = out;                       // materialized [B,U]
    float* out_proj   = out + BU;                  // projected    [B,U]
    float* out_res    = out + 2 * BU;              // residual     [B,1]
    float* out_cached = out + 2 * BU + B;          // cached_next  [B,U]
    int*   out_comm   = (int*)(out + 3 * BU + B);  // committed    [B] int32

    sdp_scan_kernel<<<dim3(B), dim3(LANES), 0, stream>>>(
        dur, spe, resid, cacheA, ulen, slen, cprev,
        out_mat, out_proj, out_res, out_cached, out_comm, U);
}